// blk_encoder_75419625718356
// MI455X (gfx1250) — compile-verified
//
#include <hip/hip_runtime.h>
#include <cmath>

// ---------------- problem constants ----------------
#define BATCH   64
#define TWOC    128
#define FDIM    512
#define KW      31
#define HALO    15          // (KW-1)/2, SAME padding
#define FT      64          // f-tile width per workgroup (conv)
#define SLAB_W  (FT + 2*HALO)   // 94
#define SLAB_P  96              // padded LDS pitch (floats)
#define PER_B   ((size_t)TWOC * FDIM)   // 65536 elements per batch

// MLP staging chunk: 128 g-rows x 64 f-cols of lin_w, LDS pitch 67 floats
#define GCHUNK  128
#define FCHUNK  64
#define WPITCH  67   // 64 data + 3 pad DWORDs (TDM pad_interval=5, pad_amount=2)

typedef __attribute__((ext_vector_type(2))) float v2f;
typedef __attribute__((ext_vector_type(8))) float v8f;
typedef __attribute__((ext_vector_type(4))) unsigned int u32x4;
typedef __attribute__((ext_vector_type(8))) int i32x8;
typedef __attribute__((ext_vector_type(4))) int i32x4;

// D(16x16,f32) = A(16x4,f32) x B(4x16,f32) + C  -- exact fp32 on the matrix pipe
__device__ __forceinline__ v8f wmma_f32(v2f a, v2f b, v8f c) {
    // (neg_a, A, neg_b, B, c_mod, C, reuse_a, reuse_b)
    return __builtin_amdgcn_wmma_f32_16x16x4_f32(false, a, false, b, (short)0, c,
                                                 false, false);
}

__device__ __forceinline__ float gelu_exact(float x) {
    return 0.5f * x * (1.0f + erff(x * 0.70710678118654752440f));
}

// ---------------- LayerNorm statistics: one block per batch ----------------
__global__ __launch_bounds__(256) void ln_stats_kernel(const float* __restrict__ x,
                                                       float* __restrict__ mean_out,
                                                       float* __restrict__ rstd_out) {
    const int b = blockIdx.x;
    const float* xb = x + (size_t)b * PER_B;
    double s = 0.0, s2 = 0.0;
    for (int i = threadIdx.x; i < (int)PER_B; i += 256) {
        float v = xb[i];
        s  += (double)v;
        s2 += (double)v * (double)v;
    }
    __shared__ double sh0[256];
    __shared__ double sh1[256];
    sh0[threadIdx.x] = s;
    sh1[threadIdx.x] = s2;
    __syncthreads();
    for (int o = 128; o > 0; o >>= 1) {
        if (threadIdx.x < (unsigned)o) {
            sh0[threadIdx.x] += sh0[threadIdx.x + o];
            sh1[threadIdx.x] += sh1[threadIdx.x + o];
        }
        __syncthreads();
    }
    if (threadIdx.x == 0) {
        double m   = sh0[0] / (double)PER_B;
        double var = sh1[0] / (double)PER_B - m * m;
        mean_out[b] = (float)m;
        rstd_out[b] = rsqrtf((float)var + 1e-5f);
    }
}

// ---------------- Conv1d as implicit GEMM via WMMA ----------------
// grid = (BATCH, FDIM/FT), block = 256 (8 waves)
// wave w owns output channels [16w, 16w+16), accumulates 4 f-subtiles of 16.
__global__ __launch_bounds__(256) void conv_wmma_kernel(
    const float* __restrict__ x0, const float* __restrict__ ln1_w,
    const float* __restrict__ ln1_b, const float* __restrict__ conv_w,
    const float* __restrict__ conv_b, const float* __restrict__ mean1,
    const float* __restrict__ rstd1, float* __restrict__ y) {
    __shared__ float slab[TWOC][SLAB_P];   // 48 KB: LN1(x0) with 15-wide halo

    const int b  = blockIdx.x;
    const int f0 = blockIdx.y * FT;
    const float mu = mean1[b];
    const float rs = rstd1[b];

    // stage LN1-normalized slab into LDS (zero pad outside [0,FDIM))
    for (int idx = threadIdx.x; idx < TWOC * SLAB_W; idx += 256) {
        const int c = idx / SLAB_W;
        const int j = idx - c * SLAB_W;
        const int f = f0 + j - HALO;
        float v = 0.0f;
        if (f >= 0 && f < FDIM) {
            const size_t off = (size_t)c * FDIM + f;
            v = (x0[(size_t)b * PER_B + off] - mu) * rs * ln1_w[off] + ln1_b[off];
        }
        slab[c][j] = v;
    }
    __syncthreads();

    const int wave  = threadIdx.x >> 5;
    const int lane  = threadIdx.x & 31;
    const int oc0   = wave * 16;
    const int m     = lane & 15;            // M (A) == N (B/D) column index
    const int rbase = (lane < 16) ? 0 : 2;  // K rows {0,2} low lanes / {1,3} via .y

    v8f acc0 = {}, acc1 = {}, acc2 = {}, acc3 = {};

    const float* wrow = conv_w + (size_t)(oc0 + m) * (TWOC * KW);  // [IC][KW] for this oc

    for (int ic0 = 0; ic0 < TWOC; ic0 += 4) {
        const float* wr0 = wrow + (ic0 + rbase) * KW;
        const float* wr1 = wrow + (ic0 + rbase + 1) * KW;
        // prefetch next ic-group's weight rows into cache (global_prefetch_b8)
        __builtin_prefetch(wr0 + 4 * KW, 0, 0);
        __builtin_prefetch(wr1 + 4 * KW, 0, 0);
        const float* s0  = &slab[ic0 + rbase][m];
        const float* s1  = &slab[ic0 + rbase + 1][m];
#pragma unroll 1
        for (int k = 0; k < KW; ++k) {
            v2f A;
            A.x = wr0[k];
            A.y = wr1[k];
            v2f B0 = { s0[k +  0], s1[k +  0] };
            v2f B1 = { s0[k + 16], s1[k + 16] };
            v2f B2 = { s0[k + 32], s1[k + 32] };
            v2f B3 = { s0[k + 48], s1[k + 48] };
            acc0 = wmma_f32(A, B0, acc0);
            acc1 = wmma_f32(A, B1, acc1);
            acc2 = wmma_f32(A, B2, acc2);
            acc3 = wmma_f32(A, B3, acc3);
        }
    }

    // write D tiles: VGPR r -> rows r (lanes 0-15) / r+8 (lanes 16-31)
    const int n     = m;
    const int mbase = (lane < 16) ? 0 : 8;
    float* yb = y + (size_t)b * PER_B + (size_t)(oc0 + mbase) * FDIM + f0 + n;
#pragma unroll
    for (int r = 0; r < 8; ++r) {
        const float bias = conv_b[oc0 + mbase + r];
        yb[(size_t)r * FDIM +  0] = acc0[r] + bias;
        yb[(size_t)r * FDIM + 16] = acc1[r] + bias;
        yb[(size_t)r * FDIM + 32] = acc2[r] + bias;
        yb[(size_t)r * FDIM + 48] = acc3[r] + bias;
    }
}

// ---------------- x1 = x0 + gelu(LN2(y)) ----------------
__global__ __launch_bounds__(256) void resid_gelu_kernel(
    const float* __restrict__ x0, const float* __restrict__ yv,
    const float* __restrict__ w, const float* __restrict__ bb,
    const float* __restrict__ mean, const float* __restrict__ rstd,
    float* __restrict__ x1) {
    const size_t i   = (size_t)blockIdx.x * 256 + threadIdx.x;
    const int    b   = (int)(i / PER_B);
    const size_t off = i - (size_t)b * PER_B;
    const float v = (yv[i] - mean[b]) * rstd[b] * w[off] + bb[off];
    x1[i] = x0[i] + gelu_exact(v);
}

// ---------------- xn = LN3(x1) ----------------
__global__ __launch_bounds__(256) void ln_apply_kernel(
    const float* __restrict__ x, const float* __restrict__ w,
    const float* __restrict__ bb, const float* __restrict__ mean,
    const float* __restrict__ rstd, float* __restrict__ out) {
    const size_t i   = (size_t)blockIdx.x * 256 + threadIdx.x;
    const int    b   = (int)(i / PER_B);
    const size_t off = i - (size_t)b * PER_B;
    out[i] = (x[i] - mean[b]) * rstd[b] * w[off] + bb[off];
}

// ---------------- MLP matmul via WMMA + TDM-staged B tiles -------------------
// out = x1 + gelu(xn @ lin_w^T + lin_b)
// grid = (BATCH, 4), block = 256 (8 waves). wave = c-tile, 8 g-tiles per wave.
// lin_w chunks [128 g x 64 f] are DMA'd into LDS by the Tensor Data Mover.
__global__ __launch_bounds__(256) void mlp_wmma_kernel(
    const float* __restrict__ xn, const float* __restrict__ lin_w,
    const float* __restrict__ lin_b, const float* __restrict__ x1,
    float* __restrict__ out) {
    __shared__ float wtile[GCHUNK][WPITCH];   // 34,304 B, pitch 67 -> conflict-free B reads

    const int b  = blockIdx.x;
    const int g0 = blockIdx.y * GCHUNK;
    const int wave  = threadIdx.x >> 5;
    const int lane  = threadIdx.x & 31;
    const int c0    = wave * 16;
    const int m     = lane & 15;
    const int rbase = (lane < 16) ? 0 : 2;

    v8f acc[8];
#pragma unroll
    for (int gt = 0; gt < 8; ++gt) acc[gt] = (v8f){};

    const float* arow = xn + (size_t)b * PER_B + (size_t)(c0 + m) * FDIM;
    const unsigned lds_base = (unsigned)(size_t)(&wtile[0][0]);

    for (int fc = 0; fc < FDIM; fc += FCHUNK) {
        __syncthreads();   // previous chunk fully consumed
        if (wave == 0) {
            // ---- Tensor DMA descriptor (ISA 8.3-8.5), 2D tile of lin_w ----
            const unsigned long long ga =
                (unsigned long long)(const void*)(lin_w + (size_t)g0 * FDIM + fc);
            u32x4 g0d;
            g0d[0] = 1u;                                   // count=1, user mode
            g0d[1] = lds_base;                             // lds_addr (bytes)
            g0d[2] = (unsigned)(ga & 0xFFFFFFFFu);         // global_addr[31:0]
            g0d[3] = (unsigned)((ga >> 32) & 0x01FFFFFFu)  // global_addr[56:32]
                   | (2u << 30);                           // type = 2 (image)
            i32x8 g1d;
            g1d[0] = (int)((2u << 16)     // data_size = 4 bytes
                   | (1u << 20)           // pad_enable
                   | (5u << 22)           // pad_interval: every 64 DWORDs
                   | (2u << 25));         // pad_amount: 3 DWORDs -> pitch 67
            g1d[1] = (int)(512u << 16);   // tensor_dim0 = 512 (bits 79:48, low half)
            g1d[2] = (int)(512u << 16);   // tensor_dim0 hi=0 | tensor_dim1 = 512 (lo)
            g1d[3] = (int)(64u << 16);    // tensor_dim1 hi=0 | tile_dim0 = 64
            g1d[4] = (int)GCHUNK;         // tile_dim1 = 128, tile_dim2 = 0
            g1d[5] = (int)FDIM;           // tensor_dim0_stride = 512 (bits 207:160)
            g1d[6] = 0;                   // dim0_stride hi | dim1_stride lo16 (=0)
            g1d[7] = (int)((FDIM * FDIM) >> 16);  // dim1_stride hi bits
            i32x4 zero4 = {0, 0, 0, 0};
            i32x8 zero8 = {0, 0, 0, 0, 0, 0, 0, 0};
            // clang-23 / therock-10.0 form: (g0, g1, g2, g3, extra, cpol)
            __builtin_amdgcn_tensor_load_to_lds(g0d, g1d, zero4, zero4, zero8, 0);
            __builtin_amdgcn_s_wait_tensorcnt(0);
        }
        __syncthreads();   // staged chunk visible to all waves

#pragma unroll 1
        for (int f0 = 0; f0 < FCHUNK; f0 += 4) {
            v2f A;
            A.x = arow[fc + f0 + rbase];
            A.y = arow[fc + f0 + rbase + 1];
#pragma unroll
            for (int gt = 0; gt < 8; ++gt) {
                const float* wr = &wtile[gt * 16 + m][f0 + rbase];
                v2f Bv = { wr[0], wr[1] };
                acc[gt] = wmma_f32(A, Bv, acc[gt]);
            }
        }
    }

    const int n     = m;
    const int mbase = (lane < 16) ? 0 : 8;
#pragma unroll
    for (int gt = 0; gt < 8; ++gt) {
        const int g = g0 + gt * 16 + n;
        const float lb = lin_b[g];
#pragma unroll
        for (int r = 0; r < 8; ++r) {
            const size_t o = (size_t)b * PER_B + (size_t)(c0 + mbase + r) * FDIM + g;
            out[o] = x1[o] + gelu_exact(acc[gt][r] + lb);
        }
    }
}

// ---------------- host-side orchestration ----------------
extern "C" void kernel_launch(void* const* d_in, const int* in_sizes, int n_in,
                              void* d_out, int out_size, void* d_ws, size_t ws_size,
                              hipStream_t stream) {
    const float* x0     = (const float*)d_in[0];
    const float* ln1_w  = (const float*)d_in[1];
    const float* ln1_b  = (const float*)d_in[2];
    const float* conv_w = (const float*)d_in[3];
    const float* conv_b = (const float*)d_in[4];
    const float* ln2_w  = (const float*)d_in[5];
    const float* ln2_b  = (const float*)d_in[6];
    const float* ln3_w  = (const float*)d_in[7];
    const float* ln3_b  = (const float*)d_in[8];
    const float* lin_w  = (const float*)d_in[9];
    const float* lin_b  = (const float*)d_in[10];
    float* out = (float*)d_out;

    const size_t N = (size_t)BATCH * PER_B;   // 4,194,304 elements
    float* buf0  = (float*)d_ws;              // holds x1 later
    float* buf1  = buf0 + N;                  // y, then xn3
    float* stats = buf1 + N;
    float* mean1 = stats + 0 * BATCH; float* rstd1 = stats + 1 * BATCH;
    float* mean2 = stats + 2 * BATCH; float* rstd2 = stats + 3 * BATCH;
    float* mean3 = stats + 4 * BATCH; float* rstd3 = stats + 5 * BATCH;

    const int ew_blocks = (int)(N / 256);     // elementwise grids

    // 1) LN1 stats of x0
    ln_stats_kernel<<<BATCH, 256, 0, stream>>>(x0, mean1, rstd1);
    // 2) y = conv(LN1(x0)) + conv_b     (WMMA implicit GEMM)
    conv_wmma_kernel<<<dim3(BATCH, FDIM / FT), 256, 0, stream>>>(
        x0, ln1_w, ln1_b, conv_w, conv_b, mean1, rstd1, buf1);
    // 3) LN2 stats of y
    ln_stats_kernel<<<BATCH, 256, 0, stream>>>(buf1, mean2, rstd2);
    // 4) x1 = x0 + gelu(LN2(y))
    resid_gelu_kernel<<<ew_blocks, 256, 0, stream>>>(x0, buf1, ln2_w, ln2_b,
                                                     mean2, rstd2, buf0);
    // 5) LN3 stats of x1
    ln_stats_kernel<<<BATCH, 256, 0, stream>>>(buf0, mean3, rstd3);
    // 6) xn3 = LN3(x1)
    ln_apply_kernel<<<ew_blocks, 256, 0, stream>>>(buf0, ln3_w, ln3_b,
                                                   mean3, rstd3, buf1);
    // 7) out = x1 + gelu(xn3 @ lin_w^T + lin_b)   (WMMA + TDM staging)
    mlp_wmma_kernel<<<dim3(BATCH, 4), 256, 0, stream>>>(buf1, lin_w, lin_b,
                                                        buf0, out);
}